// Net_86457691669085
// MI455X (gfx1250) — compile-verified
//
#include <hip/hip_runtime.h>
#include <hip/hip_bf16.h>
#include <math.h>

typedef __attribute__((ext_vector_type(2))) float v2f;
typedef __attribute__((ext_vector_type(8))) float v8f;

// ---------------------------------------------------------------------------
// Degree / norm kernels
// ---------------------------------------------------------------------------
__global__ __launch_bounds__(256) void k_deg_init(float* __restrict__ deg, int n) {
    int i = blockIdx.x * blockDim.x + threadIdx.x;
    if (i < n) deg[i] = 1.0f;  // self-loop contributes 1 to every node's degree
}

__global__ __launch_bounds__(256) void k_deg_scatter(const long long* __restrict__ dst,
                                                     float* __restrict__ deg, int nE) {
    int e = blockIdx.x * blockDim.x + threadIdx.x;
    if (e < nE) {
        long long d = __builtin_nontemporal_load(dst + e);  // streaming, no reuse
        atomicAdd(deg + d, 1.0f);
    }
}

__global__ __launch_bounds__(256) void k_dinv(float* __restrict__ deg, int n) {
    int i = blockIdx.x * blockDim.x + threadIdx.x;
    if (i < n) {
        float d = deg[i];
        deg[i] = (d > 0.0f) ? rsqrtf(d) : 0.0f;  // in-place deg -> deg^-1/2
    }
}

// ---------------------------------------------------------------------------
// Layer 1 transform: g1[node][0..15] = (dinv ⊙ x)[node] @ W1[3][16]
// (dinv folded into A rows: diagonal row-scale commutes through the GEMM)
// One V_WMMA_F32_16X16X4_F32 per wave of 16 nodes (K: 3 zero-padded to 4).
// A layout (16x4 f32): lane m=L&15, half kh=L>>4 holds A[m][2kh], A[m][2kh+1].
// D layout (16x16 f32): lane col=L&15, VGPR j holds D[j+8*(L>>4)][col].
// ---------------------------------------------------------------------------
__global__ __launch_bounds__(256) void k_gemm1_wmma(const float* __restrict__ x,
                                                    const float* __restrict__ W1,
                                                    const float* __restrict__ dinv,
                                                    float* __restrict__ g1, int n) {
    const int lane = threadIdx.x & 31;
    const int wave = (int)((blockIdx.x * blockDim.x + threadIdx.x) >> 5);
    const int base = wave * 16;
    if (base >= n) return;  // wave-uniform

    const int m    = lane & 15;
    const int kh   = lane >> 4;   // 0: K={0,1}, 1: K={2,3}
    const int col  = lane & 15;
    const int k0   = 2 * kh;      // 0 or 2 (always a valid x/W1 row)
    const int k1   = k0 + 1;      // 1 or 3 (3 is the zero-pad row)
    const bool khi = (k1 < 3);

    // B operand: branch-free via safe index + select
    v2f b;
    b.x = W1[(size_t)k0 * 16 + col];
    float wb = W1[(size_t)(khi ? k1 : 0) * 16 + col];
    b.y = khi ? wb : 0.0f;

    const int  node = base + m;
    const bool full = (base + 16 <= n);  // wave-uniform (always true for N%16==0)
    v2f a;
    if (full) {
        const float* xp = x + (size_t)node * 3;
        float di = dinv[node];
        float ax = xp[k0];
        float ay = xp[khi ? k1 : k0];  // safe index; masked below
        a.x = di * ax;
        a.y = khi ? di * ay : 0.0f;
    } else {
        bool nv = (node < n);
        const float* xp = x + (size_t)(nv ? node : 0) * 3;
        float di = nv ? dinv[node] : 0.0f;
        float ax = nv ? xp[k0] : 0.0f;
        float ay = (nv && khi) ? xp[k1] : 0.0f;
        a.x = di * ax;
        a.y = di * ay;
    }

    v8f c = {};
    c = __builtin_amdgcn_wmma_f32_16x16x4_f32(false, a, false, b, (short)0, c, false, false);

    if (full) {
        float* gp = g1 + (size_t)(base + 8 * kh) * 16 + col;
#pragma unroll
        for (int j = 0; j < 8; ++j) gp[(size_t)j * 16] = c[j];
    } else {
#pragma unroll
        for (int j = 0; j < 8; ++j) {
            int nd = base + j + 8 * kh;
            if (nd < n) g1[(size_t)nd * 16 + col] = c[j];
        }
    }
}

// ---------------------------------------------------------------------------
// Edge scatter, 16 channels: agg[dst] += g[src]  (norm already folded into g)
// ---------------------------------------------------------------------------
__global__ __launch_bounds__(256) void k_scatter16(const long long* __restrict__ src,
                                                   const long long* __restrict__ dst,
                                                   const float* __restrict__ g,
                                                   float* __restrict__ agg, int nE) {
    int e = blockIdx.x * blockDim.x + threadIdx.x;
    if (e >= nE) return;
    long long s = __builtin_nontemporal_load(src + e);  // streaming index reads
    long long d = __builtin_nontemporal_load(dst + e);
    const float4* gp = (const float4*)(g + (size_t)s * 16);
    float* ap = agg + (size_t)d * 16;
#pragma unroll
    for (int q = 0; q < 4; ++q) {
        float4 v = gp[q];
        atomicAdd(ap + 4 * q + 0, v.x);
        atomicAdd(ap + 4 * q + 1, v.y);
        atomicAdd(ap + 4 * q + 2, v.z);
        atomicAdd(ap + 4 * q + 3, v.w);
    }
}

// ---------------------------------------------------------------------------
// finalize1: h1 = relu(dinv*(agg1 + g1) + b1), written in place over agg1.
// (self-loop term dinv[i]^2 * h[i] == dinv[i] * g1[i], folded here)
// ---------------------------------------------------------------------------
__global__ __launch_bounds__(256) void k_finalize1(float* __restrict__ agg1,
                                                   const float* __restrict__ g1,
                                                   const float* __restrict__ dinv,
                                                   const float* __restrict__ b1, int n) {
    int idx = blockIdx.x * blockDim.x + threadIdx.x;
    if (idx >= n * 16) return;
    int i = idx >> 4;
    int cch = idx & 15;
    float v = dinv[i] * (agg1[idx] + g1[idx]) + b1[cch];
    agg1[idx] = v > 0.0f ? v : 0.0f;
}

// ---------------------------------------------------------------------------
// Layer 2 transform: g2[node][0..11] = (dinv ⊙ h1)[node] @ W2pad[16][12]
// Rows padded 10->12 floats (16B aligned); cols 10,11 are exact zeros since the
// corresponding B columns are zero. Four accumulating WMMAs per 16 nodes.
// ---------------------------------------------------------------------------
__global__ __launch_bounds__(256) void k_gemm2_wmma(const float* __restrict__ h1,
                                                    const float* __restrict__ W2,
                                                    const float* __restrict__ dinv,
                                                    float* __restrict__ g2, int n) {
    const int lane = threadIdx.x & 31;
    const int wave = (int)((blockIdx.x * blockDim.x + threadIdx.x) >> 5);
    const int base = wave * 16;
    if (base >= n) return;  // wave-uniform

    const int m   = lane & 15;
    const int kh  = lane >> 4;
    const int col = lane & 15;
    const int node = base + m;
    const bool full = (base + 16 <= n);
    const bool nv   = (node < n);
    const float* hp = h1 + (size_t)(nv ? node : 0) * 16;
    const float  di = nv ? dinv[node] : 0.0f;
    const bool cv = (col < 10);

    v8f c = {};
#pragma unroll
    for (int kk = 0; kk < 4; ++kk) {
        const int k0 = 4 * kk + 2 * kh;  // K pair for this lane half
        v2f a;
        a.x = di * hp[k0];
        a.y = di * hp[k0 + 1];
        v2f b;
        b.x = cv ? W2[(size_t)k0 * 10 + col] : 0.0f;
        b.y = cv ? W2[(size_t)(k0 + 1) * 10 + col] : 0.0f;
        c = __builtin_amdgcn_wmma_f32_16x16x4_f32(false, a, false, b, (short)0, c, false, false);
    }

    if (col < 12) {  // store 12-wide padded rows (cols 10,11 are zeros)
        if (full) {
            float* gp = g2 + (size_t)(base + 8 * kh) * 12 + col;
#pragma unroll
            for (int j = 0; j < 8; ++j) gp[(size_t)j * 12] = c[j];
        } else {
#pragma unroll
            for (int j = 0; j < 8; ++j) {
                int nd = base + j + 8 * kh;
                if (nd < n) g2[(size_t)nd * 12 + col] = c[j];
            }
        }
    }
}

// ---------------------------------------------------------------------------
// Edge scatter, 10 channels (rows padded to 12 -> 3x b128 gather)
// ---------------------------------------------------------------------------
__global__ __launch_bounds__(256) void k_scatter10(const long long* __restrict__ src,
                                                   const long long* __restrict__ dst,
                                                   const float* __restrict__ g,
                                                   float* __restrict__ agg, int nE) {
    int e = blockIdx.x * blockDim.x + threadIdx.x;
    if (e >= nE) return;
    long long s = __builtin_nontemporal_load(src + e);
    long long d = __builtin_nontemporal_load(dst + e);
    const float4* gp = (const float4*)(g + (size_t)s * 12);
    float4 v0 = gp[0], v1 = gp[1], v2 = gp[2];
    float* ap = agg + (size_t)d * 12;
    atomicAdd(ap + 0, v0.x); atomicAdd(ap + 1, v0.y);
    atomicAdd(ap + 2, v0.z); atomicAdd(ap + 3, v0.w);
    atomicAdd(ap + 4, v1.x); atomicAdd(ap + 5, v1.y);
    atomicAdd(ap + 6, v1.z); atomicAdd(ap + 7, v1.w);
    atomicAdd(ap + 8, v2.x); atomicAdd(ap + 9, v2.y);
}

// ---------------------------------------------------------------------------
// finalize2: z = dinv*(agg2 + g2) + b2 ; out = log_softmax(z) (10 classes)
// ---------------------------------------------------------------------------
__global__ __launch_bounds__(256) void k_finalize2(const float* __restrict__ agg2,
                                                   const float* __restrict__ g2,
                                                   const float* __restrict__ dinv,
                                                   const float* __restrict__ b2,
                                                   float* __restrict__ out, int n) {
    int i = blockIdx.x * blockDim.x + threadIdx.x;
    if (i >= n) return;
    float di = dinv[i];
    float z[10];
    float mx = -3.402823466e+38f;
#pragma unroll
    for (int c = 0; c < 10; ++c) {
        float v = di * (agg2[(size_t)i * 12 + c] + g2[(size_t)i * 12 + c]) + b2[c];
        z[c] = v;
        mx = v > mx ? v : mx;
    }
    float s = 0.0f;
#pragma unroll
    for (int c = 0; c < 10; ++c) s += expf(z[c] - mx);
    float lse = logf(s) + mx;
#pragma unroll
    for (int c = 0; c < 10; ++c) out[(size_t)i * 10 + c] = z[c] - lse;
}

// ---------------------------------------------------------------------------
// Host launcher
// ---------------------------------------------------------------------------
extern "C" void kernel_launch(void* const* d_in, const int* in_sizes, int n_in,
                              void* d_out, int out_size, void* d_ws, size_t ws_size,
                              hipStream_t stream) {
    const float*     x   = (const float*)d_in[0];
    const long long* ei  = (const long long*)d_in[1];
    const float*     W1  = (const float*)d_in[2];
    const float*     b1  = (const float*)d_in[3];
    const float*     W2  = (const float*)d_in[4];
    const float*     b2  = (const float*)d_in[5];
    float*           out = (float*)d_out;

    const int n  = in_sizes[0] / 3;   // 500000
    const int nE = in_sizes[1] / 2;   // 8000000
    const long long* srcp = ei;
    const long long* dstp = ei + nE;

    // Workspace layout (256B aligned slabs)
    auto align_up = [](size_t v) { return (v + 255) & ~(size_t)255; };
    char* ws = (char*)d_ws;
    size_t off = 0;
    float* dinv = (float*)(ws + off);  off += align_up((size_t)n * 4);        // deg -> dinv
    float* bufA = (float*)(ws + off);  off += align_up((size_t)n * 16 * 4);   // g1, then g2 (12-wide)
    float* bufB = (float*)(ws + off);  off += align_up((size_t)n * 16 * 4);   // agg1 -> h1
    float* aggC = (float*)(ws + off);  off += align_up((size_t)n * 12 * 4);   // agg2 (12-wide)
    (void)ws_size;

    const int T = 256;
    const int blkN   = (n + T - 1) / T;
    const int blkE   = (nE + T - 1) / T;
    const int waves  = (n + 15) / 16;
    const int blkW   = (waves * 32 + T - 1) / T;
    const int blkN16 = (n * 16 + T - 1) / T;

    // degrees & symmetric norm factor
    k_deg_init<<<blkN, T, 0, stream>>>(dinv, n);
    k_deg_scatter<<<blkE, T, 0, stream>>>(dstp, dinv, nE);
    k_dinv<<<blkN, T, 0, stream>>>(dinv, n);

    // layer 1
    k_gemm1_wmma<<<blkW, T, 0, stream>>>(x, W1, dinv, bufA, n);
    hipMemsetAsync(bufB, 0, (size_t)n * 16 * 4, stream);
    k_scatter16<<<blkE, T, 0, stream>>>(srcp, dstp, bufA, bufB, nE);
    k_finalize1<<<blkN16, T, 0, stream>>>(bufB, bufA, dinv, b1, n);

    // layer 2 (g2 reuses bufA; h1 lives in bufB)
    k_gemm2_wmma<<<blkW, T, 0, stream>>>(bufB, W2, dinv, bufA, n);
    hipMemsetAsync(aggC, 0, (size_t)n * 12 * 4, stream);
    k_scatter10<<<blkE, T, 0, stream>>>(srcp, dstp, bufA, aggC, nE);
    k_finalize2<<<blkN, T, 0, stream>>>(aggC, bufA, dinv, b2, out, n);
}